// MeanAggregator_17532056502285
// MI455X (gfx1250) — compile-verified
//
#include <hip/hip_runtime.h>
#include <cstdint>

// MeanAggregator: out[b, :] = mean over K=10 gathered rows of feat_table.
// Bandwidth-bound gather (≈745 MB @ 23.3 TB/s ≈ 32 us floor). Strategy:
// one wave32 per output row; indices fetched via SCALAR loads (wave-uniform,
// forced with readfirstlane so all 10 issue before a single s_wait_kmcnt);
// K async DMA copies (global -> LDS) per wave via
// global_load_async_to_lds_b128 (CDNA5 ASYNCcnt path) issued back-to-back;
// single s_wait_asynccnt 0 drain; reduce from LDS; non-temporal store.

#define K_NEIGH 10
#define DIM 128
#define WAVES_PER_BLOCK 8

typedef float v4f __attribute__((ext_vector_type(4)));

__global__ __launch_bounds__(WAVES_PER_BLOCK * 32) void mean_agg_async_kernel(
    const float* __restrict__ feat,
    const int* __restrict__ nidx,
    float* __restrict__ out,
    int nB)
{
    // 8 waves x 10 rows x 512B = 40 KB LDS per block
    __shared__ float smem[WAVES_PER_BLOCK * K_NEIGH * DIM];

    const int lane = threadIdx.x & 31;
    const int wave = threadIdx.x >> 5;
    int b = blockIdx.x * WAVES_PER_BLOCK + wave;
    if (b >= nB) return;

    // b is uniform across the wave but LLVM's divergence analysis can't see
    // it; force it into an SGPR so the index loads become s_load_b256/b64
    // (all issued at once, one s_wait_kmcnt) instead of 10 serialized
    // global_load_b32 + s_wait_loadcnt round-trips.
    b = __builtin_amdgcn_readfirstlane(b);

    float* buf = &smem[wave * (K_NEIGH * DIM)];
    const int* rowidx = nidx + b * K_NEIGH;  // uniform -> scalar loads

    // Pull all K indices into SGPRs first (single KMcnt wait).
    int rows[K_NEIGH];
#pragma unroll
    for (int k = 0; k < K_NEIGH; ++k)
        rows[k] = __builtin_amdgcn_readfirstlane(rowidx[k]);

    // Fire all K row copies asynchronously into LDS, back-to-back. Each lane
    // moves 16B; 32 lanes x 16B = one full 512B feature row per instruction.
#pragma unroll
    for (int k = 0; k < K_NEIGH; ++k) {
        const float* src = feat + (long long)rows[k] * DIM + lane * 4;
        // Low 32 bits of a generic pointer into __shared__ == LDS byte offset
        // (ISA 10.2: LDS aperture address truncation).
        unsigned lds_off = (unsigned)(uintptr_t)(buf + k * DIM + lane * 4);
        asm volatile("global_load_async_to_lds_b128 %0, %1, off"
                     :
                     : "v"(lds_off), "v"(src)
                     : "memory");
    }

    // Drain the async counter: all LDS writes for this wave are complete.
    asm volatile("s_wait_asynccnt 0" ::: "memory");

    // Reduce the K rows from LDS (ds_load_b128 per neighbor).
    v4f acc = {0.0f, 0.0f, 0.0f, 0.0f};
#pragma unroll
    for (int k = 0; k < K_NEIGH; ++k) {
        const v4f v = *(const v4f*)(buf + k * DIM + lane * 4);
        acc += v;
    }
    acc *= (1.0f / (float)K_NEIGH);

    // Streaming output, no reuse -> non-temporal store keeps L2 for the gather
    // (indices oversample ~1.3x, so the 192MB L2 has real hit potential on
    // the 512MB table; don't pollute it with the 67MB output).
    v4f* dst = (v4f*)(out + (long long)b * DIM + lane * 4);
    __builtin_nontemporal_store(acc, dst);
}

extern "C" void kernel_launch(void* const* d_in, const int* in_sizes, int n_in,
                              void* d_out, int out_size, void* d_ws, size_t ws_size,
                              hipStream_t stream)
{
    const float* feat = (const float*)d_in[0];   // [V, 128] fp32
    const int*   nidx = (const int*)d_in[1];     // [B, 10] int32 (JAX x32 default)
    float*       out  = (float*)d_out;           // [B, 128] fp32

    const int nB = in_sizes[1] / K_NEIGH;
    const int blocks = (nB + WAVES_PER_BLOCK - 1) / WAVES_PER_BLOCK;

    mean_agg_async_kernel<<<blocks, WAVES_PER_BLOCK * 32, 0, stream>>>(
        feat, nidx, out, nB);
}